// DCT_4054449128172
// MI455X (gfx1250) — compile-verified
//
#include <hip/hip_runtime.h>

typedef __attribute__((ext_vector_type(2))) float v2f;
typedef __attribute__((ext_vector_type(8))) float v8f;

#define IMG_W 1024
#define TILES_PER_ROW 64

// One 16x16 tile per wave. D = K * T * K^T via two chained 16x16x16 fp32
// products, each built from four V_WMMA_F32_16X16X4_F32 ops.
__global__ __launch_bounds__(256) void dct16_wmma_kernel(
    const float* __restrict__ x, const float* __restrict__ kmat,
    float* __restrict__ out, int nTiles) {
  __shared__ float lds[8][256];  // 1 KB per wave, wave-private
  const int lane = threadIdx.x & 31;
  const int waveInBlock = threadIdx.x >> 5;
  const int half = lane >> 4;   // 0: lanes 0-15, 1: lanes 16-31
  const int m = lane & 15;      // row (A) / col (B,C,D) index
  float* myLds = &lds[waveInBlock][0];

  // A-layout of K (== B-layout of K^T): 4 chunks of the K dimension.
  v2f kc[4];
#pragma unroll
  for (int c = 0; c < 4; ++c) {
    const int j0 = 4 * c + 2 * half;
    kc[c].x = kmat[m * 16 + j0];
    kc[c].y = kmat[m * 16 + j0 + 1];
  }

  const int wave = blockIdx.x * 8 + waveInBlock;
  const int stride = gridDim.x * 8;
  for (int t = wave; t < nTiles; t += stride) {   // wave-uniform loop
    const int trow = t >> 6;          // tile row across flattened tensor
    const int tcol = t & (TILES_PER_ROW - 1);
    const size_t base = (size_t)trow * (16 * IMG_W) + (size_t)tcol * 16;
    const float* tp = x + base;

    // ---- Step 1: C1 = K * T.  B = T loaded directly in B-layout. ----
    v8f c1 = {};
#pragma unroll
    for (int c = 0; c < 4; ++c) {
      const int j0 = 4 * c + 2 * half;  // K-dim index for this half-wave
      v2f b;
      b.x = tp[(size_t)j0 * IMG_W + m];
      b.y = tp[(size_t)(j0 + 1) * IMG_W + m];
      c1 = __builtin_amdgcn_wmma_f32_16x16x4_f32(
          false, kc[c], false, b, (short)0, c1, false, false);
    }

    // ---- Transpose C1 from C/D-layout to A-layout via LDS. ----
#pragma unroll
    for (int i = 0; i < 8; ++i) {
      myLds[(i + 8 * half) * 16 + m] = c1[i];   // lds[row*16 + col]
    }
    asm volatile("s_wait_dscnt 0" ::: "memory");

    // ---- Step 2: D = C1 * K^T.  A chunks are contiguous pairs in LDS. ----
    v8f d = {};
#pragma unroll
    for (int c = 0; c < 4; ++c) {
      const int j0 = 4 * c + 2 * half;
      v2f a;
      a.x = myLds[m * 16 + j0];
      a.y = myLds[m * 16 + j0 + 1];
      d = __builtin_amdgcn_wmma_f32_16x16x4_f32(
          false, a, false, kc[c], (short)0, d, false, false);
    }

    // ---- Store D (C/D-layout rows are lane-contiguous 64B segments). ----
    float* op = out + base;
#pragma unroll
    for (int i = 0; i < 8; ++i) {
      op[(size_t)(i + 8 * half) * IMG_W + m] = d[i];
    }
  }
}

extern "C" void kernel_launch(void* const* d_in, const int* in_sizes, int n_in,
                              void* d_out, int out_size, void* d_ws, size_t ws_size,
                              hipStream_t stream) {
  const float* x = (const float*)d_in[0];
  const float* kmat = (const float*)d_in[1];
  float* out = (float*)d_out;

  const int nTiles = in_sizes[0] / 256;        // 16x16 tiles
  int blocks = (nTiles + 7) / 8;               // 8 waves (tiles) per block
  if (blocks > 4096) blocks = 4096;            // grid-stride beyond this
  dct16_wmma_kernel<<<blocks, 256, 0, stream>>>(x, kmat, out, nTiles);
}